// PhysicsLayer_38654705664712
// MI455X (gfx1250) — compile-verified
//
#include <hip/hip_runtime.h>
#include <math.h>

// Problem constants (match reference)
#define NXK 4096   // tridiagonal system size N (grid has NXK+1 points)
#define NTK 4096   // time steps
#define NQ  1024   // query batch B
#define NTH 1024   // threads: each owns a chunk of 4 contiguous grid points

// Single-workgroup solver: 4096 sequential implicit steps. Each step is two
// constant-coefficient affine scans (Thomas fwd/bwd sweeps). All scan combine
// multipliers are time-invariant and precomputed into registers. The backward
// sweep's cross-wave inputs are expressed affinely in the forward cross-wave
// carry (C1*gB + C0 with C1 constant), so each step needs only ONE workgroup
// barrier. Interpolated output rows are emitted on the fly via a one-time
// counting sort of the B query times.
__global__ void __launch_bounds__(NTH, 1) adr_solver(
    const float* __restrict__ alpha_p, const float* __restrict__ vel_p,
    const float* __restrict__ r0_p,    const float* __restrict__ r1_p,
    const float* __restrict__ t_p,     float* __restrict__ out)
{
    const int tid  = threadIdx.x;
    const int lane = tid & 31;
    const int wv   = tid >> 5;

    // ---- LDS: one pool, phase-reused ----
    // phase A: den[4096] @0, cp[4096] @16384
    // phase B: rowstart[4097] @0, cursor[4096] @16400, order[1024] @32784,
    //          warr[1024] @36880 ; phase C: inlet[4096] @16400 (cursor dead)
    __shared__ __align__(16) unsigned char pool[41088];
    __shared__ float sF[2][32];      // fwd cross-wave carries (double buffered)
    __shared__ float sB[2][32];      // bwd cross-wave offsets (double buffered)
    __shared__ float sRed[32];
    __shared__ float sTmax;
    __shared__ float sT[NQ];         // t[] staged via async-to-LDS

    float* s_den = (float*)pool;
    float* s_cp  = (float*)(pool + 16384);

    // ---- CDNA5 async copy: t[] -> LDS (ASYNCcnt path) ----
    {
        unsigned long long ga = (unsigned long long)(const void*)(t_p + tid);
        unsigned int lo = (unsigned int)(unsigned long long)(const void*)&sT[tid];
        asm volatile("global_load_async_to_lds_b32 %0, %1, off"
                     :: "v"(lo), "v"(ga) : "memory");
    }
    asm volatile("s_wait_asynccnt 0x0" ::: "memory");
    __syncthreads();

    // ---- t_max reduction ----
    float tv = sT[tid];
    #pragma unroll
    for (int s = 16; s > 0; s >>= 1) tv = fmaxf(tv, __shfl_down(tv, s, 32));
    if (lane == 0) sRed[wv] = tv;
    __syncthreads();
    if (wv == 0) {
        float m = sRed[lane];
        #pragma unroll
        for (int s = 16; s > 0; s >>= 1) m = fmaxf(m, __shfl_down(m, s, 32));
        if (lane == 0) sTmax = m;
    }
    __syncthreads();
    const float tmax = sTmax;

    // ---- scalar parameters / matrix coefficients (f32, as reference) ----
    const float alpha = *alpha_p, vel = *vel_p, r0v = *r0_p, r1v = *r1_p;
    const float dt     = tmax / 4096.0f;
    const float inv_dt = 1.0f / dt;
    const float r_diff = alpha / 1e-4f;     // alpha / DX^2
    const float r_adv  = vel / 0.02f;       // vel / (2 DX)
    const float aa     = r_diff + r_adv;
    const float bmid   = -2.0f * r_diff - inv_dt - r1v;
    const float blast  = -(r_diff + r_adv) - inv_dt - r1v;
    const float cc     = r_diff - r_adv;

    // ---- one-time Thomas factorization (sequential continued fraction) ----
    if (tid == 0) {
        s_den[0] = 1.0f; s_cp[0] = 0.0f;
        float cpPrev = 0.0f;
        for (int i = 1; i < NXK; ++i) {
            float bi = (i == NXK - 1) ? blast : bmid;
            float ci = (i == NXK - 1) ? 0.0f  : cc;
            float den = bi - aa * cpPrev;
            float cpi = ci / den;
            s_den[i] = den; s_cp[i] = cpi; cpPrev = cpi;
        }
    }
    __syncthreads();

    // ---- per-thread constants ----
    float mco[4], gco[4], fa[4], FP[4], ba[4], RP[4];
    const float dxg = 40.96f / 4096.0f;
    #pragma unroll
    for (int k = 0; k < 4; ++k) {
        int p = 4 * tid + k;
        float invden = 1.0f / s_den[p];
        float cpv    = s_cp[p];
        if (p == 0) { fa[k] = 0.0f; mco[k] = 0.0f; gco[k] = 0.0f; }
        else {
            fa[k]  = -aa * invden;
            mco[k] = -inv_dt * invden;
            gco[k] = (r0v * ((float)p * dxg)) * invden;
        }
        ba[k] = -cpv;
    }
    FP[0] = fa[0];
    #pragma unroll
    for (int k = 1; k < 4; ++k) FP[k] = fa[k] * FP[k - 1];
    RP[3] = ba[3];
    #pragma unroll
    for (int k = 2; k >= 0; --k) RP[k] = ba[k] * RP[k + 1];

    // ---- Hillis-Steele multipliers (constants across steps) ----
    float mF[5], MF[5], mB[5], MB[5], AiF, AiB;
    {
        float A = FP[3];
        #pragma unroll
        for (int s = 0; s < 5; ++s) {
            int st = 1 << s;
            float Ap = __shfl_up(A, st, 32);
            mF[s] = A;
            if (lane >= st) A *= Ap;
        }
        AiF = A;
        if (lane == 31) sF[0][wv] = AiF;
    }
    {
        float A = RP[0];
        #pragma unroll
        for (int s = 0; s < 5; ++s) {
            int st = 1 << s;
            float Ap = __shfl_down(A, st, 32);
            mB[s] = A;
            if (lane + st < 32) A *= Ap;
        }
        AiB = A;
        if (lane == 0) sB[0][wv] = AiB;
    }
    __syncthreads();
    {
        float A = sF[0][lane];
        #pragma unroll
        for (int s = 0; s < 5; ++s) {
            int st = 1 << s;
            float Ap = __shfl_up(A, st, 32);
            MF[s] = A;
            if (lane >= st) A *= Ap;
        }
    }
    {
        float A = sB[0][lane];
        #pragma unroll
        for (int s = 0; s < 5; ++s) {
            int st = 1 << s;
            float Ap = __shfl_down(A, st, 32);
            MB[s] = A;
            if (lane + st < 32) A *= Ap;
        }
    }

    // ---- affine-fusion constants (enable single barrier per step) ----
    // carry into lane l: cin = K1*gB + K0, K1 = Ai_{l-1} (const), K0 = Bv_{l-1}
    float K1 = __shfl_up(AiF, 1, 32);
    if (lane == 0) K1 = 1.0f;
    float FPK1[4];
    #pragma unroll
    for (int k = 0; k < 4; ++k) FPK1[k] = FP[k] * K1;   // dp_k = FPK1_k*gB + dp0_k
    float RBC3 = FPK1[3];
    float RBC2 = fmaf(ba[2], RBC3, FPK1[2]);
    float RBC1 = fmaf(ba[1], RBC2, FPK1[1]);
    float RBC0 = fmaf(ba[0], RBC1, FPK1[0]);            // RB_k = RBC_k*gB + RBv_k
    float Cg;                                           // wave bwd scan of RBC0
    {
        float Bc = RBC0;
        #pragma unroll
        for (int s = 0; s < 5; ++s) {
            int st = 1 << s;
            float Bp = __shfl_down(Bc, st, 32);
            if (lane + st < 32) Bc = fmaf(mB[s], Bp, Bc);
        }
        Cg = Bc;                                        // Sb_l = Cg_l*gB + Cv_l
    }
    __syncthreads();
    if (lane == 0) sRed[wv] = Cg;                       // per-wave C1 constant
    __syncthreads();
    const float C1c = sRed[lane];

    // ---- counting sort of queries by idx0 (one time, pool reuse) ----
    int*   rowstart = (int*)pool;
    int*   cursor   = (int*)(pool + 16400);
    int*   order    = (int*)(pool + 32784);
    float* warr     = (float*)(pool + 36880);

    float tb  = sT[tid];
    float tn  = (tb / tmax) * 4096.0f;
    int   i0  = (int)floorf(tn);
    i0 = i0 < 0 ? 0 : (i0 > NTK - 1 ? NTK - 1 : i0);
    float wgt = tn - (float)i0;

    #pragma unroll
    for (int k = 0; k < 4; ++k) cursor[4 * tid + k] = 0;
    __syncthreads();
    atomicAdd(&cursor[i0], 1);
    __syncthreads();
    if (tid == 0) {
        int acc = 0; rowstart[0] = 0;
        for (int i = 0; i < NTK; ++i) { acc += cursor[i]; rowstart[i + 1] = acc; }
    }
    __syncthreads();
    #pragma unroll
    for (int k = 0; k < 4; ++k) cursor[4 * tid + k] = rowstart[4 * tid + k];
    __syncthreads();
    {
        int pos = atomicAdd(&cursor[i0], 1);
        order[pos] = tid; warr[pos] = wgt;
    }
    __syncthreads();

    // ---- hoist all inlet BC values out of the loop (cursor region is dead) --
    float* sInlet = (float*)(pool + 16400);
    const float TWO_PI = 6.28318530717958647692f;
    #pragma unroll
    for (int k = 0; k < 4; ++k) {
        int nn = 4 * tid + k;
        sInlet[nn] = 1.0f + 0.5f * sinf(TWO_PI * ((float)(nn + 1) * dt));
    }
    __syncthreads();

    // ---- time loop: 4096 steps, ONE barrier each ----
    float u0 = 1.0f, u1 = 1.0f, u2 = 1.0f, u3 = 1.0f;
    int e0 = 0;

    for (int n = 0; n < NTK; ++n) {
        const int p = n & 1;
        // rhs_i/den_i
        float bv0 = fmaf(u0, mco[0], gco[0]);
        float bv1 = fmaf(u1, mco[1], gco[1]);
        float bv2 = fmaf(u2, mco[2], gco[2]);
        float bv3 = fmaf(u3, mco[3], gco[3]);
        if (tid == 0) bv0 = sInlet[n];                  // inlet BC (precomputed)

        // forward: chunk-serial + wave scan
        float FB0 = bv0;
        float FB1 = fmaf(fa[1], FB0, bv1);
        float FB2 = fmaf(fa[2], FB1, bv2);
        float FB3 = fmaf(fa[3], FB2, bv3);
        float Bv = FB3;
        #pragma unroll
        for (int s = 0; s < 5; ++s) {
            int st = 1 << s;
            float Bp = __shfl_up(Bv, st, 32);
            if (lane >= st) Bv = fmaf(mF[s], Bp, Bv);
        }
        float K0 = __shfl_up(Bv, 1, 32);
        if (lane == 0) K0 = 0.0f;
        float dp00 = fmaf(FP[0], K0, FB0);              // dp_k at gB==0
        float dp01 = fmaf(FP[1], K0, FB1);
        float dp02 = fmaf(FP[2], K0, FB2);
        float dp03 = fmaf(FP[3], K0, FB3);
        // backward offsets (gB-independent parts), wave suffix scan
        float RBv3 = dp03;
        float RBv2 = fmaf(ba[2], RBv3, dp02);
        float RBv1 = fmaf(ba[1], RBv2, dp01);
        float RBv0 = fmaf(ba[0], RBv1, dp00);
        float Cv = RBv0;
        #pragma unroll
        for (int s = 0; s < 5; ++s) {
            int st = 1 << s;
            float Bp = __shfl_down(Cv, st, 32);
            if (lane + st < 32) Cv = fmaf(mB[s], Bp, Cv);
        }
        if (lane == 31) sF[p][wv] = Bv;                 // fwd wave carry
        if (lane == 0)  sB[p][wv] = Cv;                 // bwd wave offset C0
        __syncthreads();                                // the ONLY barrier

        // cross-wave forward scan (redundant per wave)
        float Wf = sF[p][lane];
        #pragma unroll
        for (int s = 0; s < 5; ++s) {
            int st = 1 << s;
            float Bp = __shfl_up(Wf, st, 32);
            if (lane >= st) Wf = fmaf(MF[s], Bp, Wf);
        }
        float G = __shfl_up(Wf, 1, 32);                 // carry into wave 'lane'
        if (lane == 0) G = 0.0f;
        float gB = 0.0f;                                // carry into own wave
        if (wv > 0) gB = __shfl(Wf, wv - 1, 32);
        // cross-wave backward scan on reconstructed wave summaries
        float Wb = fmaf(C1c, G, sB[p][lane]);
        #pragma unroll
        for (int s = 0; s < 5; ++s) {
            int st = 1 << s;
            float Bp = __shfl_down(Wb, st, 32);
            if (lane + st < 32) Wb = fmaf(MB[s], Bp, Wb);
        }
        float gBb = 0.0f;                               // right-carry into wave
        if (wv < 31) gBb = __shfl(Wb, wv + 1, 32);

        // reconstruct solution
        float SbSelf = fmaf(Cg, gB, Cv);                // x at chunk head, 0-carry
        float zb   = fmaf(AiB, gBb, SbSelf);            // true x at chunk head
        float cinb = __shfl_down(zb, 1, 32);
        if (lane == 31) cinb = gBb;
        float x0 = zb;
        float x1 = fmaf(RP[1], cinb, fmaf(RBC1, gB, RBv1));
        float x2 = fmaf(RP[2], cinb, fmaf(RBC2, gB, RBv2));
        float x3 = fmaf(RP[3], cinb, fmaf(RBC3, gB, RBv3));

        // emit interpolated rows for queries with idx0 == n (uniform branch)
        int e1 = rowstart[n + 1];
        if (e1 > e0) {
            for (int j = e0; j < e1; ++j) {
                int   b = order[j];
                float w = warr[j];
                float omw = 1.0f - w;
                float* o = out + (size_t)b * (NXK + 1) + 4 * tid;
                o[0] = omw * u0 + w * x0;
                o[1] = omw * u1 + w * x1;
                o[2] = omw * u2 + w * x2;
                o[3] = omw * u3 + w * x3;
                if (tid == NTH - 1) o[4] = omw * u3 + w * x3;  // Neumann col 4096
            }
        }
        e0 = e1;
        u0 = x0; u1 = x1; u2 = x2; u3 = x3;
    }
}

extern "C" void kernel_launch(void* const* d_in, const int* in_sizes, int n_in,
                              void* d_out, int out_size, void* d_ws, size_t ws_size,
                              hipStream_t stream) {
    (void)in_sizes; (void)n_in; (void)out_size; (void)d_ws; (void)ws_size;
    const float* alpha = (const float*)d_in[0];
    const float* vel   = (const float*)d_in[1];
    const float* r0    = (const float*)d_in[2];
    const float* r1    = (const float*)d_in[3];
    const float* t     = (const float*)d_in[4];
    adr_solver<<<1, NTH, 0, stream>>>(alpha, vel, r0, r1, t, (float*)d_out);
}